// ExtrudingStrokePrediction_14053132993281
// MI455X (gfx1250) — compile-verified
//
#include <hip/hip_runtime.h>
#include <hip/hip_bf16.h>
#include <math.h>

typedef __attribute__((ext_vector_type(2))) float v2f;
typedef __attribute__((ext_vector_type(8))) float v8f;

// One 16x16 f32 output tile: D += A(16xK, rows in LDS fb) * B(KxN row-major, ldb)
// using V_WMMA_F32_16X16X4_F32 over the K dimension (K multiple of 4).
__device__ __forceinline__ v8f wmma_k(const float* fb, int fstride,
                                      const float* W, int ldb,
                                      int ntile, int K, v8f acc, int lane) {
  const int r    = lane & 15;          // A row (M) / B col within tile (N)
  const int koff = (lane >> 4) << 1;   // lanes 0-15 -> K+0,K+1 ; lanes 16-31 -> K+2,K+3
  const int cc   = (ntile << 4) + r;
#pragma unroll
  for (int k = 0; k < K; k += 4) {
    v2f a, b;
    a.x = fb[r * fstride + k + koff];
    a.y = fb[r * fstride + k + koff + 1];
    b.x = W[(k + koff) * ldb + cc];
    b.y = W[(k + koff + 1) * ldb + cc];
    acc = __builtin_amdgcn_wmma_f32_16x16x4_f32(false, a, false, b,
                                                (short)0, acc, false, false);
  }
  return acc;
}

// -------- workspace init: zero sums/counts, set max buffer to -inf bits ------
__global__ __launch_bounds__(256) void k_init(float* z, long nz, int* mx, long nm) {
  long i = (long)blockIdx.x * 256 + threadIdx.x;
  if (i < nz) z[i] = 0.f;
  if (i < nm) mx[i] = (int)0xFF800000u;   // -inf
}

// -------- y_sc = (x * (1 + id)) @ W_sc   (bias added later, gated) ----------
__global__ __launch_bounds__(256) void k_ysc(const float* __restrict__ x,
                                             const float* __restrict__ id,
                                             const float* __restrict__ Wsc,
                                             float* __restrict__ y, long NS) {
  __shared__ float sW[32 * 32];
  __shared__ float sFb[8][16 * 68];
  __shared__ float sId[8][16];
  const int tid = threadIdx.x;
  for (int i = tid; i < 1024; i += 256) sW[i] = Wsc[i];
  const int wave = tid >> 5, lane = tid & 31;
  const long base = ((long)blockIdx.x * 8 + wave) * 16;
  const long NSm1 = NS - 1;
  float* fb = sFb[wave];
  if (lane < 16) {
    long n = base + lane; if (n > NSm1) n = NSm1;
    sId[wave][lane] = id[n];
  }
  __syncthreads();
  for (int row = 0; row < 16; ++row) {
    long n = base + row; if (n > NSm1) n = NSm1;
    fb[row * 68 + lane] = x[n * 32 + lane] * (1.f + sId[wave][row]);
  }
  __syncthreads();
  v8f a0 = {}, a1 = {};
  a0 = wmma_k(fb, 68, sW, 32, 0, 32, a0, lane);
  a1 = wmma_k(fb, 68, sW, 32, 1, 32, a1, lane);
#pragma unroll
  for (int v = 0; v < 8; ++v) {
    int row = (lane < 16) ? v : v + 8;
    long n = base + row;
    if (n < NS) {
      y[n * 32 + (lane & 15)]      = a0[v];
      y[n * 32 + 16 + (lane & 15)] = a1[v];
    }
  }
}

// -------- edge scatter-sum of gated stroke features + count -----------------
__global__ __launch_bounds__(256) void k_edge_sum_stroke(const int* __restrict__ e, long E,
    const float* __restrict__ x, const float* __restrict__ id,
    float* __restrict__ s, float* __restrict__ c) {
  long i = (long)blockIdx.x * 256 + threadIdx.x;
  if (i >= E * 32) return;
  long ei = i >> 5; int col = (int)(i & 31);
  int src = e[ei]; int dst = e[E + ei];
  float v = x[(long)src * 32 + col] * (1.f + id[src]);
  atomicAdd(&s[(long)dst * 32 + col], v);
  if (col == 0) atomicAdd(&c[dst], 1.f);
}

// -------- edge scatter-sum of raw (brep) features + count -------------------
__global__ __launch_bounds__(256) void k_edge_sum_raw(const int* __restrict__ e, long E,
    const float* __restrict__ x, float* __restrict__ s, float* __restrict__ c) {
  long i = (long)blockIdx.x * 256 + threadIdx.x;
  if (i >= E * 32) return;
  long ei = i >> 5; int col = (int)(i & 31);
  int src = e[ei]; int dst = e[E + ei];
  atomicAdd(&s[(long)dst * 32 + col], x[(long)src * 32 + col]);
  if (col == 0) atomicAdd(&c[dst], 1.f);
}

// -------- edge scatter-max via sign-split int atomics (stores real f32 bits)
__global__ __launch_bounds__(256) void k_edge_max(const int* __restrict__ e, long E,
    const float* __restrict__ y, int* __restrict__ mx) {
  long i = (long)blockIdx.x * 256 + threadIdx.x;
  if (i >= E * 32) return;
  long ei = i >> 5; int col = (int)(i & 31);
  int src = e[ei]; int dst = e[E + ei];
  int b = __float_as_int(y[(long)src * 32 + col]);
  int* p = &mx[(long)dst * 32 + col];
  if (b >= 0) atomicMax(p, b);
  else        atomicMin((unsigned int*)p, (unsigned int)b);
}

// -------- per-node: 3 relation transforms + combine + MLP head --------------
__global__ __launch_bounds__(128) void k_node(
    const float* __restrict__ s_int, const float* __restrict__ c_int,
    const float* __restrict__ s_tp,  const float* __restrict__ c_tp,
    const float* __restrict__ s_rb,  const float* __restrict__ c_rb,
    const int*   __restrict__ mx,
    const float* __restrict__ x, const float* __restrict__ id,
    const float* __restrict__ Wint, const float* __restrict__ bint,
    const float* __restrict__ Wtp,  const float* __restrict__ btp,
    const float* __restrict__ Wrb,  const float* __restrict__ brb,
    const float* __restrict__ bsc,
    const float* __restrict__ Wl,   const float* __restrict__ bl,
    const float* __restrict__ Wd1,  const float* __restrict__ bd1,
    const float* __restrict__ Wd2,  const float* __restrict__ bd2,
    float* __restrict__ out, long NS)
{
  __shared__ float sWint[1024], sWtp[1024], sWrb[1024];
  __shared__ float sWl[2048], sWd1[4096], sWd2[64];
  __shared__ float sBint[32], sBtp[32], sBrb[32], sBsc[32], sBl[64], sBd1[64];
  __shared__ float sBd2;
  __shared__ float sFb[4][16 * 68];
  __shared__ float sCint[4][16], sCtp[4][16], sCrb[4][16], sId[4][16];

  const int tid = threadIdx.x;
  for (int i = tid; i < 1024; i += 128) { sWint[i]=Wint[i]; sWtp[i]=Wtp[i]; sWrb[i]=Wrb[i]; }
  for (int i = tid; i < 2048; i += 128) sWl[i]  = Wl[i];
  for (int i = tid; i < 4096; i += 128) sWd1[i] = Wd1[i];
  if (tid < 64) { sWd2[tid] = Wd2[tid]; sBl[tid] = bl[tid]; sBd1[tid] = bd1[tid]; }
  if (tid < 32) { sBint[tid]=bint[tid]; sBtp[tid]=btp[tid]; sBrb[tid]=brb[tid]; sBsc[tid]=bsc[tid]; }
  if (tid == 0) sBd2 = bd2[0];

  const int wave = tid >> 5, lane = tid & 31;
  const long base = ((long)blockIdx.x * 4 + wave) * 16;
  const long NSm1 = NS - 1;
  float* fb = sFb[wave];
  if (lane < 16) {
    long n = base + lane; if (n > NSm1) n = NSm1;
    sCint[wave][lane] = c_int[n];
    sCtp[wave][lane]  = c_tp[n];
    sCrb[wave][lane]  = c_rb[n];
    sId[wave][lane]   = id[n];
  }
  __syncthreads();

  v8f acc0 = {}, acc1 = {};

  // ---- relation 1: mean(intersects) @ W_int
  for (int row = 0; row < 16; ++row) {
    long n = base + row; if (n > NSm1) n = NSm1;
    float c = sCint[wave][row];
    fb[row * 68 + lane] = (c > 0.f) ? s_int[n * 32 + lane] / c : 0.f;
  }
  __syncthreads();
  acc0 = wmma_k(fb, 68, sWint, 32, 0, 32, acc0, lane);
  acc1 = wmma_k(fb, 68, sWint, 32, 1, 32, acc1, lane);
  __syncthreads();

  // ---- relation 2: sum(temp_prev) @ W_tp
  for (int row = 0; row < 16; ++row) {
    long n = base + row; if (n > NSm1) n = NSm1;
    fb[row * 68 + lane] = s_tp[n * 32 + lane];
  }
  __syncthreads();
  acc0 = wmma_k(fb, 68, sWtp, 32, 0, 32, acc0, lane);
  acc1 = wmma_k(fb, 68, sWtp, 32, 1, 32, acc1, lane);
  __syncthreads();

  // ---- relation 3: mean(repr, brep src) @ W_rb
  for (int row = 0; row < 16; ++row) {
    long n = base + row; if (n > NSm1) n = NSm1;
    float c = sCrb[wave][row];
    fb[row * 68 + lane] = (c > 0.f) ? s_rb[n * 32 + lane] / c : 0.f;
  }
  __syncthreads();
  acc0 = wmma_k(fb, 68, sWrb, 32, 0, 32, acc0, lane);
  acc1 = wmma_k(fb, 68, sWrb, 32, 1, 32, acc1, lane);
  __syncthreads();

  // ---- biases (gated), max relation, relu, residual -> stroke_out in fb
#pragma unroll
  for (int n = 0; n < 2; ++n) {
    v8f acc = n ? acc1 : acc0;
#pragma unroll
    for (int v = 0; v < 8; ++v) {
      int row = (lane < 16) ? v : v + 8;
      int col = n * 16 + (lane & 15);
      long node = base + row; long nc = node > NSm1 ? NSm1 : node;
      float s = acc[v];
      if (sCint[wave][row] > 0.f) s += sBint[col];
      s += sCtp[wave][row] * sBtp[col];
      if (sCrb[wave][row] > 0.f) s += sBrb[col];
      int mb = mx[nc * 32 + col];
      if (mb != (int)0xFF800000u) s += __int_as_float(mb) + sBsc[col];
      float xsv = x[nc * 32 + col] * (1.f + sId[wave][row]);
      fb[row * 68 + col] = fmaxf(s, 0.f) + xsv;
    }
  }
  __syncthreads();

  // ---- feats = stroke_out @ W_local + b_local   (32 -> 64)
  v8f f[4] = {};
#pragma unroll
  for (int n = 0; n < 4; ++n) f[n] = wmma_k(fb, 68, sWl, 64, n, 32, f[n], lane);
  __syncthreads();
#pragma unroll
  for (int n = 0; n < 4; ++n)
#pragma unroll
    for (int v = 0; v < 8; ++v) {
      int row = (lane < 16) ? v : v + 8;
      int col = n * 16 + (lane & 15);
      fb[row * 68 + col] = f[n][v] + sBl[col];
    }
  __syncthreads();

  // ---- h = relu(feats @ W_d1 + b_d1)   (64 -> 64)
  v8f h[4] = {};
#pragma unroll
  for (int n = 0; n < 4; ++n) h[n] = wmma_k(fb, 68, sWd1, 64, n, 64, h[n], lane);
  __syncthreads();
#pragma unroll
  for (int n = 0; n < 4; ++n)
#pragma unroll
    for (int v = 0; v < 8; ++v) {
      int row = (lane < 16) ? v : v + 8;
      int col = n * 16 + (lane & 15);
      fb[row * 68 + col] = fmaxf(h[n][v] + sBd1[col], 0.f);
    }
  __syncthreads();

  // ---- out = sigmoid(h @ W_d2 + b_d2)   (64 -> 1)
  if (lane < 16) {
    long node = base + lane;
    if (node < NS) {
      float z = sBd2;
#pragma unroll
      for (int k = 0; k < 64; ++k) z += fb[lane * 68 + k] * sWd2[k];
      out[node] = 1.f / (1.f + __expf(-z));
    }
  }
}

extern "C" void kernel_launch(void* const* d_in, const int* in_sizes, int n_in,
                              void* d_out, int out_size, void* d_ws, size_t ws_size,
                              hipStream_t stream) {
  const float* x_stroke = (const float*)d_in[0];
  const float* x_brep   = (const float*)d_in[1];
  const float* sid      = (const float*)d_in[2];
  const int*   e_int    = (const int*)d_in[3];
  const int*   e_tp     = (const int*)d_in[4];
  const int*   e_rb     = (const int*)d_in[5];
  const int*   e_sc     = (const int*)d_in[6];
  const float* Wint = (const float*)d_in[7],  *bint = (const float*)d_in[8];
  const float* Wtp  = (const float*)d_in[9],  *btp  = (const float*)d_in[10];
  const float* Wrb  = (const float*)d_in[11], *brb  = (const float*)d_in[12];
  const float* Wsc  = (const float*)d_in[13], *bsc  = (const float*)d_in[14];
  const float* Wl   = (const float*)d_in[15], *bl   = (const float*)d_in[16];
  const float* Wd1  = (const float*)d_in[17], *bd1  = (const float*)d_in[18];
  const float* Wd2  = (const float*)d_in[19], *bd2  = (const float*)d_in[20];
  float* out = (float*)d_out;

  long NS   = in_sizes[0] / 32;
  long Ei   = in_sizes[3] / 2;
  long Et   = in_sizes[4] / 2;
  long Er   = in_sizes[5] / 2;
  long Ec   = in_sizes[6] / 2;
  long NS32 = NS * 32;

  float* ws    = (float*)d_ws;
  float* s_int = ws;
  float* c_int = s_int + NS32;
  float* s_tp  = c_int + NS;
  float* c_tp  = s_tp + NS32;
  float* s_rb  = c_tp + NS;
  float* c_rb  = s_rb + NS32;
  long   nz    = 3 * (NS32 + NS);
  int*   mx    = (int*)(c_rb + NS);
  float* ysc   = (float*)(mx + NS32);

  long ninit = nz > NS32 ? nz : NS32;
  k_init<<<(unsigned)((ninit + 255) / 256), 256, 0, stream>>>(ws, nz, mx, NS32);

  long tiles = (NS + 15) / 16;
  k_ysc<<<(unsigned)((tiles + 7) / 8), 256, 0, stream>>>(x_stroke, sid, Wsc, ysc, NS);

  k_edge_sum_stroke<<<(unsigned)((Ei * 32 + 255) / 256), 256, 0, stream>>>(e_int, Ei, x_stroke, sid, s_int, c_int);
  k_edge_sum_stroke<<<(unsigned)((Et * 32 + 255) / 256), 256, 0, stream>>>(e_tp,  Et, x_stroke, sid, s_tp,  c_tp);
  k_edge_sum_raw  <<<(unsigned)((Er * 32 + 255) / 256), 256, 0, stream>>>(e_rb,  Er, x_brep,  s_rb, c_rb);
  k_edge_max      <<<(unsigned)((Ec * 32 + 255) / 256), 256, 0, stream>>>(e_sc,  Ec, ysc, mx);

  k_node<<<(unsigned)((tiles + 3) / 4), 128, 0, stream>>>(
      s_int, c_int, s_tp, c_tp, s_rb, c_rb, mx, x_stroke, sid,
      Wint, bint, Wtp, btp, Wrb, brb, bsc, Wl, bl, Wd1, bd1, Wd2, bd2, out, NS);
}